// ParticleFlow_17222818857123
// MI455X (gfx1250) — compile-verified
//
#include <hip/hip_runtime.h>
#include <math.h>

// ---------------------------------------------------------------------------
// RealNVP-style flow, fused forward+inverse, fp32 WMMA (V_WMMA_F32_16X16X4_F32)
// One wave32 = 16 points. Hidden 16x16 layers run on the matrix pipe with
// exact fp32 math (required for the invertibility round-trip).
// ---------------------------------------------------------------------------

typedef float v2f __attribute__((ext_vector_type(2)));
typedef float v8f __attribute__((ext_vector_type(8)));

#define NBLOCKS 6
#define NPTS    131072
#define NBATCH  4

// Param flattening assumption (dict order of setup_inputs):
//   d_in[0] = xyz
//   per block (35 arrays): logs, bias, W_perm,
//     c1F{W0,W1,W2,b0,b1,b2,Wout,bout}, c1G{...}, c2F{...}, c2G{...}
struct FlowParams { const float* p[NBLOCKS * 35]; };

#define UNIT(P, base)                                                      \
  (P).p[(base) + 0], (P).p[(base) + 1], (P).p[(base) + 2],                 \
  (P).p[(base) + 3], (P).p[(base) + 4], (P).p[(base) + 5],                 \
  (P).p[(base) + 6], (P).p[(base) + 7]

__device__ __forceinline__ v8f wmma_f32_k4(v2f a, v2f b, v8f c) {
  // 8 args: (neg_a, A, neg_b, B, c_mod, C, reuse_a, reuse_b)
  return __builtin_amdgcn_wmma_f32_16x16x4_f32(false, a, false, b,
                                               (short)0, c, false, false);
}

__device__ __forceinline__ float softclamp(float s) {
  // CLAMP * (2/pi) * atan(s / CLAMP), CLAMP = 1.9
  return 1.9f * 0.636619772367581343f * atanf(s * (1.0f / 1.9f));
}

// Activations a[8] hold a 16x16 f32 matrix in WMMA C/D layout:
//   lane l, reg v  <->  hidden m = v + 8*(l/16), point n = l%16.

// Input layer: a = relu(W0 @ [x0;x1] + b0). W0 is 16x2, K padded 2->4.
__device__ __forceinline__ void layer_in(const float* __restrict__ W0,
                                         const float* __restrict__ b0,
                                         float x0, float x1,
                                         float a[8], unsigned lane) {
  const bool lo  = lane < 16u;
  const int  n   = lane & 15;
  const int  hi8 = (int)(lane >> 4) << 3;
  v2f A, B;
  A.x = lo ? W0[n * 2 + 0] : 0.0f;   // A: K = {0,1} lo-half, {2,3}=pad hi-half
  A.y = lo ? W0[n * 2 + 1] : 0.0f;
  B.x = lo ? x0 : 0.0f;              // B: K rows 0,1 real; 2,3 pad
  B.y = lo ? x1 : 0.0f;
  v8f c;
#pragma unroll
  for (int v = 0; v < 8; ++v) c[v] = b0[hi8 + v];
  c = wmma_f32_k4(A, B, c);
#pragma unroll
  for (int v = 0; v < 8; ++v) a[v] = fmaxf(c[v], 0.0f);
}

// Hidden layer: a = relu(W @ a + b). W is 16x16 row-major [out][in].
__device__ __forceinline__ void layer_hidden(const float* __restrict__ W,
                                             const float* __restrict__ b,
                                             float a[8], unsigned lane) {
  const bool lo  = lane < 16u;
  const int  n   = lane & 15;
  const int  hi2 = (int)(lane >> 4) << 1;
  const int  hi8 = (int)(lane >> 4) << 3;
  // Rebuild B operand (K x N, chunked K=4) from D-layout: 8 half-wave swaps.
  float s0 = __shfl_xor(a[0], 16), s1 = __shfl_xor(a[1], 16);
  float s2 = __shfl_xor(a[2], 16), s3 = __shfl_xor(a[3], 16);
  float s4 = __shfl_xor(a[4], 16), s5 = __shfl_xor(a[5], 16);
  float s6 = __shfl_xor(a[6], 16), s7 = __shfl_xor(a[7], 16);
  v2f B0, B1, B2, B3;
  B0.x = lo ? a[0] : s2;  B0.y = lo ? a[1] : s3;   // K = 0..3
  B1.x = lo ? a[4] : s6;  B1.y = lo ? a[5] : s7;   // K = 4..7
  B2.x = lo ? s0 : a[2];  B2.y = lo ? s1 : a[3];   // K = 8..11
  B3.x = lo ? s4 : a[6];  B3.y = lo ? s5 : a[7];   // K = 12..15
  // A operand: weight K-slices, lane holds W[n][4c + 2*hi + {0,1}]
  const float* wrow = W + n * 16 + hi2;
  v2f A0, A1, A2, A3;
  A0.x = wrow[0];  A0.y = wrow[1];
  A1.x = wrow[4];  A1.y = wrow[5];
  A2.x = wrow[8];  A2.y = wrow[9];
  A3.x = wrow[12]; A3.y = wrow[13];
  v8f c;
#pragma unroll
  for (int v = 0; v < 8; ++v) c[v] = b[hi8 + v];
  c = wmma_f32_k4(A0, B0, c);
  c = wmma_f32_k4(A1, B1, c);
  c = wmma_f32_k4(A2, B2, c);
  c = wmma_f32_k4(A3, B3, c);
#pragma unroll
  for (int v = 0; v < 8; ++v) a[v] = fmaxf(c[v], 0.0f);
}

// Output layer: scalar = Wout . h + bout, replicated in both half-lanes.
__device__ __forceinline__ float layer_out(const float* __restrict__ Wout,
                                           const float* __restrict__ bout,
                                           const float a[8], unsigned lane) {
  const int hi8 = (int)(lane >> 4) << 3;
  float p = 0.0f;
#pragma unroll
  for (int v = 0; v < 8; ++v) p = fmaf(a[v], Wout[hi8 + v], p);
  p += __shfl_xor(p, 16);
  return p + bout[0];
}

__device__ __forceinline__ float eval_unit(const float* W0, const float* W1,
                                           const float* W2, const float* b0,
                                           const float* b1, const float* b2,
                                           const float* Wout, const float* bout,
                                           float x0, float x1, unsigned lane) {
  float a[8];
  layer_in(W0, b0, x0, x1, a, lane);
  layer_hidden(W1, b1, a, lane);
  layer_hidden(W2, b2, a, lane);
  return layer_out(Wout, bout, a, lane);
}

// Sum a per-point value (replicated in both half-lanes) over the 16 points.
__device__ __forceinline__ float tile_sum(float v, unsigned lane) {
  v = (lane < 16u) ? v : 0.0f;
#pragma unroll
  for (int off = 16; off >= 1; off >>= 1) v += __shfl_xor(v, off);
  return v;
}

__global__ __launch_bounds__(256)
void flow_fused_kernel(const float* __restrict__ xyz, FlowParams P,
                       float* __restrict__ out, float* __restrict__ ws) {
  __shared__ float sAcc[25];   // [0..11] fwd log_s sums, [12..23] inv, [24] z*z
  if (threadIdx.x < 25) sAcc[threadIdx.x] = 0.0f;
  __syncthreads();

  const unsigned lane = threadIdx.x & 31u;
  const unsigned wave = threadIdx.x >> 5;
  const unsigned tile = blockIdx.x * 8u + wave;
  const unsigned pt   = tile * 16u + (lane & 15u);  // same point in both halves
  const int      bat  = (int)(pt / NPTS);           // uniform per workgroup

  float x0 = xyz[(size_t)pt * 3 + 0];
  float x1 = xyz[(size_t)pt * 3 + 1];
  float x2 = xyz[(size_t)pt * 3 + 2];

  // ---------------- forward: x -> z ----------------
  for (int blk = 0; blk < NBLOCKS; ++blk) {
    const int base = blk * 35;
    const float* logs = P.p[base + 0];
    const float* bias = P.p[base + 1];
    const float* W    = P.p[base + 2];
    x0 = x0 * expf(logs[0]) + bias[0];
    x1 = x1 * expf(logs[1]) + bias[1];
    x2 = x2 * expf(logs[2]) + bias[2];
    float y0 = x0 * W[0] + x1 * W[3] + x2 * W[6];   // x @ W_perm
    float y1 = x0 * W[1] + x1 * W[4] + x2 * W[7];
    float y2 = x0 * W[2] + x1 * W[5] + x2 * W[8];
    // coupling 1 (c1F @ base+3, c1G @ base+11)
    float ls1 = softclamp(eval_unit(UNIT(P, base + 3),  y0, y1, lane));
    float t1  =           eval_unit(UNIT(P, base + 11), y0, y1, lane);
    y2 = y2 * expf(ls1) + t1;
    {
      float r = tile_sum(ls1, lane);
      if (lane == 0) atomicAdd(&sAcc[blk * 2 + 0], r);
    }
    // channel reverse then coupling 2 (c2F @ base+19, c2G @ base+27)
    float r0 = y2, r1 = y1, r2 = y0;
    float ls2 = softclamp(eval_unit(UNIT(P, base + 19), r0, r1, lane));
    float t2  =           eval_unit(UNIT(P, base + 27), r0, r1, lane);
    r2 = r2 * expf(ls2) + t2;
    {
      float r = tile_sum(ls2, lane);
      if (lane == 0) atomicAdd(&sAcc[blk * 2 + 1], r);
    }
    x0 = r0; x1 = r1; x2 = r2;
  }

  if (lane < 16u) {                    // write z
    size_t o = (size_t)pt * 3;
    out[o] = x0; out[o + 1] = x1; out[o + 2] = x2;
  }
  {
    float r = tile_sum(x0 * x0 + x1 * x1 + x2 * x2, lane);
    if (lane == 0) atomicAdd(&sAcc[24], r);
  }

  // ---------------- inverse: z -> x_hat ----------------
  for (int blk = NBLOCKS - 1; blk >= 0; --blk) {
    const int base = blk * 35;
    const float* logs = P.p[base + 0];
    const float* bias = P.p[base + 1];
    const float* W    = P.p[base + 2];
    // coupling 2 inverse
    float ls2 = softclamp(eval_unit(UNIT(P, base + 19), x0, x1, lane));
    float t2  =           eval_unit(UNIT(P, base + 27), x0, x1, lane);
    x2 = (x2 - t2) * expf(-ls2);
    {
      float r = tile_sum(ls2, lane);
      if (lane == 0) atomicAdd(&sAcc[12 + blk * 2 + 1], r);
    }
    // channel reverse then coupling 1 inverse
    float r0 = x2, r1 = x1, r2 = x0;
    float ls1 = softclamp(eval_unit(UNIT(P, base + 3),  r0, r1, lane));
    float t1  =           eval_unit(UNIT(P, base + 11), r0, r1, lane);
    r2 = (r2 - t1) * expf(-ls1);
    {
      float r = tile_sum(ls1, lane);
      if (lane == 0) atomicAdd(&sAcc[12 + blk * 2 + 0], r);
    }
    // multiply by inv(W_perm) via adjugate (uniform scalar math)
    float m00 = W[0], m01 = W[1], m02 = W[2];
    float m10 = W[3], m11 = W[4], m12 = W[5];
    float m20 = W[6], m21 = W[7], m22 = W[8];
    float c00 = m11 * m22 - m12 * m21;
    float c01 = m12 * m20 - m10 * m22;
    float c02 = m10 * m21 - m11 * m20;
    float c10 = m02 * m21 - m01 * m22;
    float c11 = m00 * m22 - m02 * m20;
    float c12 = m01 * m20 - m00 * m21;
    float c20 = m01 * m12 - m02 * m11;
    float c21 = m02 * m10 - m00 * m12;
    float c22 = m00 * m11 - m01 * m10;
    float det  = m00 * c00 + m01 * c01 + m02 * c02;
    float rdet = 1.0f / det;
    float u0 = (r0 * c00 + r1 * c01 + r2 * c02) * rdet;
    float u1 = (r0 * c10 + r1 * c11 + r2 * c12) * rdet;
    float u2 = (r0 * c20 + r1 * c21 + r2 * c22) * rdet;
    x0 = (u0 - bias[0]) * expf(-logs[0]);
    x1 = (u1 - bias[1]) * expf(-logs[1]);
    x2 = (u2 - bias[2]) * expf(-logs[2]);
  }

  if (lane < 16u) {                    // write x_hat
    size_t o = (size_t)(NBATCH * NPTS) * 3 + (size_t)pt * 3;
    out[o] = x0; out[o + 1] = x1; out[o + 2] = x2;
  }

  __syncthreads();
  if (threadIdx.x < 25) atomicAdd(&ws[bat * 25 + threadIdx.x], sAcc[threadIdx.x]);
}

__global__ void flow_zero_ws(float* __restrict__ ws) {
  unsigned i = threadIdx.x;
  if (i < NBATCH * 25) ws[i] = 0.0f;
}

__global__ void flow_finalize_kernel(FlowParams P, const float* __restrict__ ws,
                                     float* __restrict__ out) {
  int t = (int)threadIdx.x;
  if (t >= NBATCH) return;
  const float Nf = (float)NPTS;
  float J = 0.0f, Aabs = 0.0f, Jr = 0.0f;
  for (int blk = 0; blk < NBLOCKS; ++blk) {
    const int base = blk * 35;
    const float* logs = P.p[base + 0];
    const float* W    = P.p[base + 2];
    float sl = logs[0] + logs[1] + logs[2];
    float m00 = W[0], m01 = W[1], m02 = W[2];
    float m10 = W[3], m11 = W[4], m12 = W[5];
    float m20 = W[6], m21 = W[7], m22 = W[8];
    float det = m00 * (m11 * m22 - m12 * m21)
              - m01 * (m10 * m22 - m12 * m20)
              + m02 * (m10 * m21 - m11 * m20);
    float labs = logf(fabsf(det));
    float ldf = Nf * (sl + labs) + ws[t * 25 + blk * 2 + 0]
                                 + ws[t * 25 + blk * 2 + 1];
    J += ldf;
    Aabs += fabsf(ldf);
    float ldi = -(ws[t * 25 + 12 + blk * 2 + 0] +
                  ws[t * 25 + 12 + blk * 2 + 1]) - Nf * (labs + sl);
    Jr += ldi;
  }
  float zz   = ws[t * 25 + 24];
  float logp = -0.5f * (Nf * 3.0f) * logf(6.283185307179586f) - 0.5f * zz;
  size_t so = (size_t)2 * NBATCH * NPTS * 3;
  out[so + 0 * NBATCH + t] = J;      // log_det_J
  out[so + 1 * NBATCH + t] = logp;   // logp_z
  out[so + 2 * NBATCH + t] = Jr;     // log_det_J_rev
  out[so + 3 * NBATCH + t] = Aabs;   // log_det_abs
}

extern "C" void kernel_launch(void* const* d_in, const int* in_sizes, int n_in,
                              void* d_out, int out_size, void* d_ws, size_t ws_size,
                              hipStream_t stream) {
  (void)in_sizes; (void)n_in; (void)out_size; (void)ws_size;
  FlowParams P;
  for (int i = 0; i < NBLOCKS * 35; ++i) P.p[i] = (const float*)d_in[1 + i];
  const float* xyz = (const float*)d_in[0];
  float* out = (float*)d_out;
  float* ws  = (float*)d_ws;

  flow_zero_ws<<<dim3(1), dim3(128), 0, stream>>>(ws);
  flow_fused_kernel<<<dim3((NBATCH * NPTS) / 128), dim3(256), 0, stream>>>(
      xyz, P, out, ws);
  flow_finalize_kernel<<<dim3(1), dim3(32), 0, stream>>>(P, ws, out);
}